// QuantumGate_12704513262262
// MI455X (gfx1250) — compile-verified
//
#include <hip/hip_runtime.h>

// Quantum product-state simulator, collapsed form.
//
// Reference applies only single-qubit gates to |0..0>, so the state is a
// tensor product:  out[b,i] = prod_q p[b][q][bit_q(i)]  with
//   p0 = (cy*cx)^2 + (sy*sx)^2,  p1 = (sy*cx)^2 + (cy*sx)^2
// (Rz contributes only a phase -> cancels under |.|^2).
//
// Stage 1: per-(batch,qubit) probability pairs (4096 threads, trivial).
// Stage 2: per batch, build Alo[256] (qubits 0..7) and Bhi[256] (qubits 8..15)
//          in LDS, then out[b] = Bhi (outer) Alo via V_WMMA_F32_16X16X4_F32
//          rank-1 tiles (A: 16x4 with only K=0 column live; B: 4x16 with only
//          K=0 row live). 67 MB of fp32 stores -> pure HBM-bound (~3 us floor).

#define NQ    16
#define BATCH 256
#define DIM   65536

typedef __attribute__((ext_vector_type(2))) float v2f;
typedef __attribute__((ext_vector_type(8))) float v8f;

__global__ __launch_bounds__(256) void qgate_probs_kernel(
    const float* __restrict__ emb,      // [256, 768]
    const float* __restrict__ rot,      // [16, 3]
    float2* __restrict__ probs)         // [256, 16] -> (p0, p1)
{
    int idx = blockIdx.x * 256 + threadIdx.x;   // b*16 + q
    if (idx >= BATCH * NQ) return;
    int b = idx >> 4;
    int q = idx & 15;

    float ax = emb[b * 768 + q * 3 + 0] + rot[q * 3 + 0];
    float ay = emb[b * 768 + q * 3 + 1] + rot[q * 3 + 1];
    // rz angle only contributes a phase; |amp|^2 is independent of it.

    float sx, cx, sy, cy;
    sincosf(ax * 0.5f, &sx, &cx);
    sincosf(ay * 0.5f, &sy, &cy);

    float r0 = cy * cx, i0 = sy * sx;   // component 0 of (Ry Rx) e0
    float r1 = sy * cx, i1 = cy * sx;   // component 1
    probs[idx] = make_float2(r0 * r0 + i0 * i0, r1 * r1 + i1 * i1);
}

__global__ __launch_bounds__(256) void qstate_expand_kernel(
    const float2* __restrict__ probs,   // [256, 16]
    float* __restrict__ out)            // [256, 65536]
{
    const int b    = blockIdx.x;
    const int t    = threadIdx.x;       // 0..255
    const int wave = t >> 5;            // 0..7
    const int lane = t & 31;

    __shared__ float2 pq[NQ];
    __shared__ float  Alo[256];         // product over qubits 0..7
    __shared__ float  Bhi[256];         // product over qubits 8..15

    if (t < NQ) pq[t] = probs[b * NQ + t];
    __syncthreads();

    float a = 1.0f, h = 1.0f;
#pragma unroll
    for (int q = 0; q < 8; ++q) {
        float2 lo = pq[q];
        float2 hi = pq[q + 8];
        a *= ((t >> q) & 1) ? lo.y : lo.x;
        h *= ((t >> q) & 1) ? hi.y : hi.x;
    }
    Alo[t] = a;
    Bhi[t] = h;
    __syncthreads();

    float* __restrict__ outb = out + (size_t)b * DIM;
    const int halfsel = (lane < 16) ? 1 : 0;   // lanes 0-15 carry K=0 data
    const int l16     = lane & 15;

    // 256 16x16 tiles of the 256x256 outer product; 32 tiles per wave.
#pragma unroll 4
    for (int k = 0; k < 32; ++k) {
        int tt = wave * 32 + k;
        int ti = tt >> 4;               // hi-tile (rows / M)
        int tj = tt & 15;               // lo-tile (cols / N)

        // A (16x4, M-major): lanes 0-15 VGPR0 hold A[M=lane, K=0].
        float av = Bhi[ti * 16 + l16];
        // B (4x16, rows across lanes): lanes 0-15 VGPR0 hold B[K=0, N=lane].
        float bv = Alo[tj * 16 + l16];

        v2f A, B;
        A.x = halfsel ? av : 0.0f;  A.y = 0.0f;
        B.x = halfsel ? bv : 0.0f;  B.y = 0.0f;
        v8f C = {};

        // D = A x B + 0  -> rank-1 16x16 tile: D[m,n] = Bhi[...m] * Alo[...n]
        v8f D = __builtin_amdgcn_wmma_f32_16x16x4_f32(
            /*neg_a=*/false, A, /*neg_b=*/false, B,
            /*c_mod=*/(short)0, C, /*reuse_a=*/false, /*reuse_b=*/false);

        // D layout: VGPR r, lanes 0-15 -> (M=r, N=lane); lanes 16-31 -> (M=r+8).
        int m0 = (lane < 16) ? 0 : 8;
        int n  = tj * 16 + l16;
#pragma unroll
        for (int r = 0; r < 8; ++r) {
            int hi_row = ti * 16 + m0 + r;
            outb[(size_t)hi_row * 256 + n] = D[r];
        }
    }
}

extern "C" void kernel_launch(void* const* d_in, const int* in_sizes, int n_in,
                              void* d_out, int out_size, void* d_ws, size_t ws_size,
                              hipStream_t stream) {
    const float* emb = (const float*)d_in[0];   // [256, 768]
    const float* rot = (const float*)d_in[1];   // [16, 3]
    float* out = (float*)d_out;                 // [256, 65536]
    float2* probs = (float2*)d_ws;              // 256*16 float2 = 32 KB scratch

    qgate_probs_kernel<<<(BATCH * NQ + 255) / 256, 256, 0, stream>>>(emb, rot, probs);
    qstate_expand_kernel<<<BATCH, 256, 0, stream>>>(probs, out);
}